// SimplifiedSSM_35622458753225
// MI455X (gfx1250) — compile-verified
//
#include <hip/hip_runtime.h>
#include <hip/hip_bf16.h>
#include <math.h>

typedef __attribute__((ext_vector_type(2))) float v2f;
typedef __attribute__((ext_vector_type(4))) float v4f;
typedef __attribute__((ext_vector_type(8))) float v8f;
typedef __attribute__((ext_vector_type(4))) int   v4i;

typedef __attribute__((address_space(1))) v4i as1_v4i;   // global int4
typedef __attribute__((address_space(3))) v4i as3_v4i;   // LDS int4

#define D_MODEL 768
#define D_STATE 16
#define D_INNER 1536
#define BATCH   2
#define SEQ     2048
#define NTOK    (BATCH * SEQ)      /* 4096 */
#define NSEG    16
#define SEGLEN  (SEQ / NSEG)       /* 128 */
#define KC      16                 /* GEMM K-chunk staged in LDS */

#if __has_builtin(__builtin_amdgcn_global_load_async_to_lds_b128) && \
    __has_builtin(__builtin_amdgcn_s_wait_asynccnt)
#define HAVE_ASYNC_LDS 1
#else
#define HAVE_ASYNC_LDS 0
#endif

__device__ __forceinline__ float silu_f(float x) {
    return x * (1.0f / (1.0f + __expf(-x)));
}

__device__ __forceinline__ v8f wmma4(v2f a, v2f b, v8f c) {
    return __builtin_amdgcn_wmma_f32_16x16x4_f32(
        false, a, false, b, (short)0, c, false, false);
}

// ---------------------------------------------------------------------------
// Register-blocked fp32 WMMA GEMM with LDS-staged B panel.
//   C[M,N] = A[M,K] * B[K,N]  (row-major).  Block = 4 waves = 128x64 C tile;
//   each wave owns 32x64 (2 M-tiles x 4 N-tiles, 8 v8f accumulators).
// All 4 waves share the same 64 B-columns, so the KCx64 B chunk is staged
// into LDS once per block via GLOBAL_LOAD_ASYNC_TO_LDS_B128 (ASYNCcnt),
// double-buffered so the DMA of chunk c+1 overlaps the 32 WMMAs of chunk c.
// A fragments are wave-private streams: direct global b64 + prefetch.
// Fragment layouts (wave32):
//   A (16x4): lane l<16 -> M=l,K={0,1};  lane>=16 -> M=l-16,K={2,3}
//   B (4x16): lane l<16 -> N=l,K={0,1};  lane>=16 -> N=l-16,K={2,3}
//   C:        VGPR r, lanes 0-15 -> M=r, lanes 16-31 -> M=r+8, N=lane&15
// Requires M % 128 == 0, N % 64 == 0, K % KC == 0.
// mode 0: store acc.  mode 1: store softplus(acc + bias[col]).
// ---------------------------------------------------------------------------
__global__ __launch_bounds__(128)
void wmma_gemm_f32_blk(const float* __restrict__ A,
                       const float* __restrict__ Bm,
                       float* __restrict__ C,
                       int M, int N, int K,
                       const float* __restrict__ bias, int mode)
{
    __shared__ float Bs[2][KC][64];              // 2 x 4KB double buffer

    const int lane = threadIdx.x & 31;
    const int wave = threadIdx.x >> 5;
    const int m    = lane & 15;
    const int koff = (lane >> 4) << 1;           // 0 (lanes 0-15) / 2 (16-31)

    const int row0 = blockIdx.y * 128 + wave * 32;   // wave's 2 M-tiles
    const int col0 = blockIdx.x * 64;                // block's 4 N-tiles

    const float* aPtr0 = A + (size_t)(row0 + m) * K + koff;
    const float* aPtr1 = aPtr0 + (size_t)16 * K;

    // B staging: chunk = KC*64 = 1024 floats = 256 float4; 2 float4 per thread.
    const int tid = threadIdx.x;
    const int f0  = tid * 2, f1 = tid * 2 + 1;
    const int sr0 = f0 >> 4, sc0 = (f0 & 15) << 2;   // row, col-of-float4
    const int sr1 = f1 >> 4, sc1 = (f1 & 15) << 2;
    const float* gB0 = Bm + (size_t)sr0 * N + col0 + sc0;
    const float* gB1 = Bm + (size_t)sr1 * N + col0 + sc1;
    const size_t chunkStride = (size_t)KC * N;       // advance per chunk

    const int nc = K / KC;

    // ---- stage chunk 0 into buffer 0 ----
#if HAVE_ASYNC_LDS
    __builtin_amdgcn_global_load_async_to_lds_b128(
        (as1_v4i*)gB0, (as3_v4i*)&Bs[0][sr0][sc0], 0, 0);
    __builtin_amdgcn_global_load_async_to_lds_b128(
        (as1_v4i*)gB1, (as3_v4i*)&Bs[0][sr1][sc1], 0, 0);
#else
    *(v4f*)&Bs[0][sr0][sc0] = *(const v4f*)gB0;
    *(v4f*)&Bs[0][sr1][sc1] = *(const v4f*)gB1;
#endif

    v8f acc[2][4] = {};

    for (int c = 0; c < nc; ++c) {
        const int buf = c & 1;

        // ---- kick off DMA of chunk c+1 into the other buffer ----
        if (c + 1 < nc) {
            const float* g0 = gB0 + (size_t)(c + 1) * chunkStride;
            const float* g1 = gB1 + (size_t)(c + 1) * chunkStride;
#if HAVE_ASYNC_LDS
            __builtin_amdgcn_global_load_async_to_lds_b128(
                (as1_v4i*)g0, (as3_v4i*)&Bs[1 - buf][sr0][sc0], 0, 0);
            __builtin_amdgcn_global_load_async_to_lds_b128(
                (as1_v4i*)g1, (as3_v4i*)&Bs[1 - buf][sr1][sc1], 0, 0);
            __builtin_amdgcn_s_wait_asynccnt(2);   // chunk c's 2 loads done
#else
            *(v4f*)&Bs[1 - buf][sr0][sc0] = *(const v4f*)g0;
            *(v4f*)&Bs[1 - buf][sr1][sc1] = *(const v4f*)g1;
#endif
        } else {
#if HAVE_ASYNC_LDS
            __builtin_amdgcn_s_wait_asynccnt(0);   // drain
#endif
        }
        __syncthreads();                           // chunk c visible to all waves

        const int kbase = c * KC;
        __builtin_prefetch(aPtr0 + kbase + 64, 0, 1);   // global_prefetch_b8
        __builtin_prefetch(aPtr1 + kbase + 64, 0, 1);

        #pragma unroll
        for (int kk = 0; kk < KC; kk += 4) {
            v2f a0 = *(const v2f*)(aPtr0 + kbase + kk);
            v2f a1 = *(const v2f*)(aPtr1 + kbase + kk);
            const float* lb0 = &Bs[buf][kk + koff][m];       // ds_load_b32 x4
            const float* lb1 = &Bs[buf][kk + koff + 1][m];
            #pragma unroll
            for (int j = 0; j < 4; ++j) {
                v2f b;
                b.x = lb0[j * 16];
                b.y = lb1[j * 16];
                acc[0][j] = wmma4(a0, b, acc[0][j]);
                acc[1][j] = wmma4(a1, b, acc[1][j]);
            }
        }
        __syncthreads();    // done reading buf before it is re-staged
    }

    const int rowHalf = (lane >> 4) << 3;          // 0 or 8
    #pragma unroll
    for (int i = 0; i < 2; ++i) {
        const int rbase = row0 + i * 16 + rowHalf;
        #pragma unroll
        for (int j = 0; j < 4; ++j) {
            const int colC = col0 + j * 16 + m;
            const float bi = (mode == 1) ? bias[colC] : 0.0f;
            #pragma unroll
            for (int r = 0; r < 8; ++r) {
                float v = acc[i][j][r];
                if (mode == 1) {
                    v += bi;
                    v = (v > 20.0f) ? v : log1pf(__expf(v));   // softplus
                }
                C[(size_t)(rbase + r) * N + colC] = v;
            }
        }
    }
}

// ---------------------------------------------------------------------------
// Simple 1-tile-per-wave WMMA GEMM for narrow N (the N=32 xproj GEMM).
// ---------------------------------------------------------------------------
__global__ __launch_bounds__(128)
void wmma_gemm_f32_narrow(const float* __restrict__ A,
                          const float* __restrict__ Bm,
                          float* __restrict__ C,
                          int M, int N, int K)
{
    const int lane  = threadIdx.x & 31;
    const int wave  = threadIdx.x >> 5;
    const int mTile = blockIdx.y * 4 + wave;
    const int nTile = blockIdx.x;
    const int row0  = mTile << 4;
    const int col0  = nTile << 4;
    const int m     = lane & 15;
    const int koff  = (lane >> 4) << 1;

    const float* aPtr = A  + (size_t)(row0 + m) * K + koff;
    const float* bPtr = Bm + (size_t)koff * N + col0 + m;

    v8f acc = {};
    for (int k = 0; k < K; k += 4) {
        v2f a = *(const v2f*)(aPtr + k);
        v2f b;
        b.x = bPtr[(size_t)k * N];
        b.y = bPtr[(size_t)(k + 1) * N];
        acc = wmma4(a, b, acc);
    }

    const int colC    = col0 + m;
    const int rowBase = row0 + ((lane >> 4) << 3);
    #pragma unroll
    for (int r = 0; r < 8; ++r)
        C[(size_t)(rowBase + r) * N + colC] = acc[r];
}

// ---------------------------------------------------------------------------
// Depthwise conv1d (k=3, pad=1, cross-correlation) + bias + SiLU.
// xz is [NTOK, 2*D_INNER]; x_c = cols [0,D_INNER). Output xc [NTOK, D_INNER].
// ---------------------------------------------------------------------------
__global__ void conv_silu_kernel(const float* __restrict__ xz,
                                 const float* __restrict__ conv_w,
                                 const float* __restrict__ conv_b,
                                 float* __restrict__ xc)
{
    int idx = blockIdx.x * blockDim.x + threadIdx.x;
    if (idx >= NTOK * D_INNER) return;
    int d   = idx % D_INNER;
    int tok = idx / D_INNER;
    int t   = tok % SEQ;

    const float w0 = conv_w[d * 3 + 0];
    const float w1 = conv_w[d * 3 + 1];
    const float w2 = conv_w[d * 3 + 2];
    const size_t rs = 2 * D_INNER;
    const float* base = xz + (size_t)tok * rs + d;

    float acc = conv_b[d] + w1 * base[0];
    if (t > 0)       acc += w0 * base[-(long)rs];
    if (t < SEQ - 1) acc += w2 * base[rs];
    xc[idx] = silu_f(acc);
}

// ---------------------------------------------------------------------------
// Segmented diagonal scan, pass 1: per (b, seg, d) compute
//   P[n] = prod_t exp(delta*A[n]),  q[n] = state after segment from h=0.
// ---------------------------------------------------------------------------
__global__ void scan_pass1(const float* __restrict__ delta,
                           const float* __restrict__ xc,
                           const float* __restrict__ bc,
                           const float* __restrict__ A_log,
                           float* __restrict__ Pseg,
                           float* __restrict__ Qseg)
{
    int tid = blockIdx.x * blockDim.x + threadIdx.x;
    if (tid >= BATCH * NSEG * D_INNER) return;
    int d = tid % D_INNER;
    int s = (tid / D_INNER) % NSEG;
    int b = tid / (D_INNER * NSEG);

    float Ar[D_STATE], h[D_STATE], P[D_STATE];
    #pragma unroll
    for (int n = 0; n < D_STATE; ++n) {
        Ar[n] = -__expf(A_log[d * D_STATE + n]);
        h[n]  = 0.0f;
        P[n]  = 1.0f;
    }

    const int t0 = s * SEGLEN;
    for (int t = t0; t < t0 + SEGLEN; ++t) {
        size_t tok = (size_t)b * SEQ + t;
        float dl = delta[tok * D_INNER + d];
        float xv = xc[tok * D_INNER + d];
        float dx = dl * xv;
        const float* brow = bc + tok * (2 * D_STATE);
        #pragma unroll
        for (int n = 0; n < D_STATE; ++n) {
            float dA = __expf(dl * Ar[n]);
            h[n] = dA * h[n] + dx * brow[n];
            P[n] *= dA;
        }
    }
    size_t o = ((size_t)(b * NSEG + s) * D_INNER + d) * D_STATE;
    #pragma unroll
    for (int n = 0; n < D_STATE; ++n) { Pseg[o + n] = P[n]; Qseg[o + n] = h[n]; }
}

// ---------------------------------------------------------------------------
// Pass 2: sequential prefix over the 16 segments (elementwise affine
// composition h <- P*h + q). One thread per (b, d), 16-vector in registers.
// ---------------------------------------------------------------------------
__global__ void scan_pass2(const float* __restrict__ Pseg,
                           const float* __restrict__ Qseg,
                           float* __restrict__ Hinit)
{
    int tid = blockIdx.x * blockDim.x + threadIdx.x;
    if (tid >= BATCH * D_INNER) return;
    int d = tid % D_INNER;
    int b = tid / D_INNER;

    float h[D_STATE];
    #pragma unroll
    for (int n = 0; n < D_STATE; ++n) h[n] = 0.0f;

    for (int s = 0; s < NSEG; ++s) {
        size_t o = ((size_t)(b * NSEG + s) * D_INNER + d) * D_STATE;
        #pragma unroll
        for (int n = 0; n < D_STATE; ++n) Hinit[o + n] = h[n];
        #pragma unroll
        for (int n = 0; n < D_STATE; ++n) h[n] = Pseg[o + n] * h[n] + Qseg[o + n];
    }
}

// ---------------------------------------------------------------------------
// Pass 3: replay each segment from its correct initial state, emit
//   ypre = (y + D*xc) * silu(z)   ready for the output GEMM.
// ---------------------------------------------------------------------------
__global__ void scan_pass3(const float* __restrict__ delta,
                           const float* __restrict__ xc,
                           const float* __restrict__ bc,
                           const float* __restrict__ A_log,
                           const float* __restrict__ Dskip,
                           const float* __restrict__ xz,
                           const float* __restrict__ Hinit,
                           float* __restrict__ ypre)
{
    int tid = blockIdx.x * blockDim.x + threadIdx.x;
    if (tid >= BATCH * NSEG * D_INNER) return;
    int d = tid % D_INNER;
    int s = (tid / D_INNER) % NSEG;
    int b = tid / (D_INNER * NSEG);

    float Ar[D_STATE], h[D_STATE];
    size_t o = ((size_t)(b * NSEG + s) * D_INNER + d) * D_STATE;
    #pragma unroll
    for (int n = 0; n < D_STATE; ++n) {
        Ar[n] = -__expf(A_log[d * D_STATE + n]);
        h[n]  = Hinit[o + n];
    }
    const float Dd = Dskip[d];

    const int t0 = s * SEGLEN;
    for (int t = t0; t < t0 + SEGLEN; ++t) {
        size_t tok = (size_t)b * SEQ + t;
        float dl = delta[tok * D_INNER + d];
        float xv = xc[tok * D_INNER + d];
        float dx = dl * xv;
        const float* brow = bc + tok * (2 * D_STATE);
        const float* crow = brow + D_STATE;
        float y = 0.0f;
        #pragma unroll
        for (int n = 0; n < D_STATE; ++n) {
            float dA = __expf(dl * Ar[n]);
            h[n] = dA * h[n] + dx * brow[n];
            y += h[n] * crow[n];
        }
        float z = xz[tok * (size_t)(2 * D_INNER) + D_INNER + d];
        ypre[tok * D_INNER + d] = (y + Dd * xv) * silu_f(z);
    }
}

// ---------------------------------------------------------------------------
extern "C" void kernel_launch(void* const* d_in, const int* in_sizes, int n_in,
                              void* d_out, int out_size, void* d_ws, size_t ws_size,
                              hipStream_t stream)
{
    (void)in_sizes; (void)n_in; (void)out_size; (void)ws_size;

    const float* x       = (const float*)d_in[0];
    const float* in_w    = (const float*)d_in[1];
    const float* conv_w  = (const float*)d_in[2];
    const float* conv_b  = (const float*)d_in[3];
    const float* xproj_w = (const float*)d_in[4];
    const float* dt_w    = (const float*)d_in[5];
    const float* dt_b    = (const float*)d_in[6];
    const float* A_log   = (const float*)d_in[7];
    const float* Dskip   = (const float*)d_in[8];
    const float* out_w   = (const float*)d_in[9];
    float* out = (float*)d_out;

    float* ws    = (float*)d_ws;
    float* xz    = ws; ws += (size_t)NTOK * 2 * D_INNER;               // 12.6M
    float* xc    = ws; ws += (size_t)NTOK * D_INNER;                   // 6.3M
    float* delta = ws; ws += (size_t)NTOK * D_INNER;                   // 6.3M
    float* bcbuf = ws; ws += (size_t)NTOK * 2 * D_STATE;               // 131K
    float* Pseg  = ws; ws += (size_t)BATCH * NSEG * D_INNER * D_STATE; // 786K
    float* Qseg  = ws; ws += (size_t)BATCH * NSEG * D_INNER * D_STATE; // 786K
    float* Hinit = ws; ws += (size_t)BATCH * NSEG * D_INNER * D_STATE; // 786K
    float* ypre  = ws;                                                 // 6.3M

    const dim3 gblk(128);  // 4 waves per block (wave32)

    // 1) xz = x @ in_w                     [4096 x 3072], K=768
    wmma_gemm_f32_blk<<<dim3(2 * D_INNER / 64, NTOK / 128), gblk, 0, stream>>>(
        x, in_w, xz, NTOK, 2 * D_INNER, D_MODEL, nullptr, 0);

    // 2) depthwise conv + SiLU -> xc
    {
        int nel = NTOK * D_INNER;
        conv_silu_kernel<<<(nel + 255) / 256, 256, 0, stream>>>(xz, conv_w, conv_b, xc);
    }

    // 3) delta = softplus(xc @ dt_w + dt_b)  [4096 x 1536], K=1536
    wmma_gemm_f32_blk<<<dim3(D_INNER / 64, NTOK / 128), gblk, 0, stream>>>(
        xc, dt_w, delta, NTOK, D_INNER, D_INNER, dt_b, 1);

    // 4) [Bs|Cs] = xc @ xproj_w              [4096 x 32], K=1536
    wmma_gemm_f32_narrow<<<dim3(2 * D_STATE / 16, NTOK / 64), gblk, 0, stream>>>(
        xc, xproj_w, bcbuf, NTOK, 2 * D_STATE, D_INNER);

    // 5-7) segmented diagonal scan (16 segments of 128 steps)
    {
        int np1 = BATCH * NSEG * D_INNER;   // 49152
        int np2 = BATCH * D_INNER;          // 3072
        scan_pass1<<<(np1 + 255) / 256, 256, 0, stream>>>(delta, xc, bcbuf, A_log, Pseg, Qseg);
        scan_pass2<<<(np2 + 255) / 256, 256, 0, stream>>>(Pseg, Qseg, Hinit);
        scan_pass3<<<(np1 + 255) / 256, 256, 0, stream>>>(delta, xc, bcbuf, A_log, Dskip,
                                                          xz, Hinit, ypre);
    }

    // 8) out = ypre @ out_w                  [4096 x 768], K=1536
    wmma_gemm_f32_blk<<<dim3(D_MODEL / 64, NTOK / 128), gblk, 0, stream>>>(
        ypre, out_w, out, NTOK, D_MODEL, D_INNER, nullptr, 0);
}